// CovidsV3_15633680957940
// MI455X (gfx1250) — compile-verified
//
#include <hip/hip_runtime.h>

typedef float v2f __attribute__((ext_vector_type(2)));
typedef float v8f __attribute__((ext_vector_type(8)));

#define DEV __device__ __forceinline__

// D = A(16x4) * B(4x16) + C, fp32 WMMA (CDNA5 V_WMMA_F32_16X16X4_F32)
DEV v8f wmma4(v2f a, v2f b, v8f c) {
  return __builtin_amdgcn_wmma_f32_16x16x4_f32(false, a, false, b, (short)0, c, false, false);
}

DEV float frelu(float v) { return v > 0.0f ? v : 0.0f; }

// ---------------------------------------------------------------------------
// Kernel A: per (b,m) block.  x -> center -> Sigma -> U=relu(Sigma*WU1+b)*WU2+b
//           -> x_norm -> center -> Sigma_norm.  Also dumps Sigma & group means
//           to workspace for kernels B / C1.
// grid = B*M = 8192 blocks, 256 threads (8 waves)
// ---------------------------------------------------------------------------
__global__ __launch_bounds__(256)
void kA(const float* __restrict__ x, const float* __restrict__ WU1,
        const float* __restrict__ bU1, const float* __restrict__ WU2,
        const float* __restrict__ bU2,
        float* __restrict__ out_xnorm, float* __restrict__ out_sign,
        float* __restrict__ ws_sigma, float* __restrict__ ws_mean)
{
  const int blk = blockIdx.x;
  const int b   = blk >> 8;
  const int m   = blk & 255;
  const int tid = threadIdx.x;
  const int lane = tid & 31, wid = tid >> 5;
  const int half = lane >> 4, li = lane & 15, k2 = half * 2;

  __shared__ float s_xc[16 * 65];   // centered x tile (rows 12..15 zero)
  __shared__ float s_xn[16 * 65];   // x_norm tile
  __shared__ float s_S [64 * 65];   // Sigma
  __shared__ float s_Ht[64 * 65];   // H chunk (64 cols of 256)
  __shared__ float s_U [64 * 65];   // U
  __shared__ float s_mu[64];

  // ---- load + center over T=12 ----
  const float* xb = x + (size_t)(b * 3072 + m * 12) * 64;
  for (int i = tid; i < 16 * 64; i += 256) {
    int t = i >> 6, d = i & 63;
    s_xc[t * 65 + d] = (t < 12) ? xb[t * 64 + d] : 0.0f;
  }
  __syncthreads();
  if (tid < 64) {
    float s = 0.0f;
    for (int t = 0; t < 12; ++t) s += s_xc[t * 65 + tid];
    float mu = s * (1.0f / 12.0f);
    s_mu[tid] = mu;
    ws_mean[(size_t)blk * 64 + tid] = mu;
    for (int t = 0; t < 12; ++t) s_xc[t * 65 + tid] -= mu;
  }
  __syncthreads();

  // ---- Sigma = Xc^T Xc / 11 : 4x4 tiles of 16x16, 2 per wave, K=12 ----
  const float inv11 = 1.0f / 11.0f;
  float* wsS = ws_sigma + (size_t)blk * 4096;
  for (int t8 = 0; t8 < 2; ++t8) {
    int tile = wid * 2 + t8, tm = tile >> 2, tn = tile & 3;
    v8f acc = {};
    for (int kb = 0; kb < 12; kb += 4) {
      v2f a, bb;
      a.x  = s_xc[(kb + k2 + 0) * 65 + tm * 16 + li];
      a.y  = s_xc[(kb + k2 + 1) * 65 + tm * 16 + li];
      bb.x = s_xc[(kb + k2 + 0) * 65 + tn * 16 + li];
      bb.y = s_xc[(kb + k2 + 1) * 65 + tn * 16 + li];
      acc = wmma4(a, bb, acc);
    }
    for (int r = 0; r < 8; ++r) {
      int row = tm * 16 + r + 8 * half, col = tn * 16 + li;
      float v = acc[r] * inv11;
      s_S[row * 65 + col] = v;
      wsS[row * 64 + col] = v;
    }
  }
  __syncthreads();

  // ---- U = relu(Sigma@WU1 + bU1) @ WU2  (chunked over H=256 in 4x64) ----
  v8f accU0 = {}, accU1 = {};
  const int ut0 = wid * 2, ut1 = wid * 2 + 1;
  for (int c = 0; c < 4; ++c) {
    const int cH = c * 64;
    // H chunk: 16 tiles, 2 per wave, K=64
    for (int t8 = 0; t8 < 2; ++t8) {
      int tile = wid * 2 + t8, tm = tile >> 2, tn = tile & 3;
      v8f acc = {};
      for (int kb = 0; kb < 64; kb += 4) {
        v2f a, bb;
        a.x  = s_S[(tm * 16 + li) * 65 + kb + k2 + 0];
        a.y  = s_S[(tm * 16 + li) * 65 + kb + k2 + 1];
        bb.x = WU1[(kb + k2 + 0) * 256 + cH + tn * 16 + li];
        bb.y = WU1[(kb + k2 + 1) * 256 + cH + tn * 16 + li];
        acc = wmma4(a, bb, acc);
      }
      float bias = bU1[cH + tn * 16 + li];
      for (int r = 0; r < 8; ++r)
        s_Ht[(tm * 16 + r + 8 * half) * 65 + tn * 16 + li] = frelu(acc[r] + bias);
    }
    __syncthreads();
    // U accumulation: each wave owns 2 tiles, K=64 over this chunk
    {
      int tm0 = ut0 >> 2, tn0 = ut0 & 3, tm1 = ut1 >> 2, tn1 = ut1 & 3;
      for (int kb = 0; kb < 64; kb += 4) {
        v2f a0, b0, a1, b1;
        a0.x = s_Ht[(tm0 * 16 + li) * 65 + kb + k2 + 0];
        a0.y = s_Ht[(tm0 * 16 + li) * 65 + kb + k2 + 1];
        b0.x = WU2[(cH + kb + k2 + 0) * 64 + tn0 * 16 + li];
        b0.y = WU2[(cH + kb + k2 + 1) * 64 + tn0 * 16 + li];
        accU0 = wmma4(a0, b0, accU0);
        a1.x = s_Ht[(tm1 * 16 + li) * 65 + kb + k2 + 0];
        a1.y = s_Ht[(tm1 * 16 + li) * 65 + kb + k2 + 1];
        b1.x = WU2[(cH + kb + k2 + 0) * 64 + tn1 * 16 + li];
        b1.y = WU2[(cH + kb + k2 + 1) * 64 + tn1 * 16 + li];
        accU1 = wmma4(a1, b1, accU1);
      }
    }
    __syncthreads();
  }
  {
    int tm0 = ut0 >> 2, tn0 = ut0 & 3, tm1 = ut1 >> 2, tn1 = ut1 & 3;
    float bias0 = bU2[tn0 * 16 + li], bias1 = bU2[tn1 * 16 + li];
    for (int r = 0; r < 8; ++r) {
      s_U[(tm0 * 16 + r + 8 * half) * 65 + tn0 * 16 + li] = accU0[r] + bias0;
      s_U[(tm1 * 16 + r + 8 * half) * 65 + tn1 * 16 + li] = accU1[r] + bias1;
    }
  }
  __syncthreads();

  // ---- x_norm = Xc @ U^T : 1x4 tiles (rows padded to 16), waves 0..3 ----
  if (wid < 4) {
    int tn = wid;
    v8f acc = {};
    for (int kb = 0; kb < 64; kb += 4) {
      v2f a, bb;
      a.x  = s_xc[li * 65 + kb + k2 + 0];          // A[t][n] = xc
      a.y  = s_xc[li * 65 + kb + k2 + 1];
      bb.x = s_U[(tn * 16 + li) * 65 + kb + k2 + 0]; // B[n][v] = U[v][n]
      bb.y = s_U[(tn * 16 + li) * 65 + kb + k2 + 1];
      acc = wmma4(a, bb, acc);
    }
    float* xo = out_xnorm + (size_t)(b * 3072 + m * 12) * 64;
    for (int r = 0; r < 8; ++r) {
      int t = r + 8 * half, col = tn * 16 + li;
      s_xn[t * 65 + col] = acc[r];                 // rows 12..15 are exact 0
      if (t < 12) xo[t * 64 + col] = acc[r];
    }
  }
  __syncthreads();

  // ---- center x_norm over T ----
  if (tid < 64) {
    float s = 0.0f;
    for (int t = 0; t < 12; ++t) s += s_xn[t * 65 + tid];
    float mu = s * (1.0f / 12.0f);
    for (int t = 0; t < 12; ++t) s_xn[t * 65 + tid] -= mu;
  }
  __syncthreads();

  // ---- Sigma_norm = Xnc^T Xnc / 11 ----
  float* so = out_sign + (size_t)blk * 4096;
  for (int t8 = 0; t8 < 2; ++t8) {
    int tile = wid * 2 + t8, tm = tile >> 2, tn = tile & 3;
    v8f acc = {};
    for (int kb = 0; kb < 12; kb += 4) {
      v2f a, bb;
      a.x  = s_xn[(kb + k2 + 0) * 65 + tm * 16 + li];
      a.y  = s_xn[(kb + k2 + 1) * 65 + tm * 16 + li];
      bb.x = s_xn[(kb + k2 + 0) * 65 + tn * 16 + li];
      bb.y = s_xn[(kb + k2 + 1) * 65 + tn * 16 + li];
      acc = wmma4(a, bb, acc);
    }
    for (int r = 0; r < 8; ++r)
      so[(tm * 16 + r + 8 * half) * 64 + tn * 16 + li] = acc[r] * inv11;
  }
}

// ---------------------------------------------------------------------------
// Kernel B: U2 = relu(s_perm @ Wu2a + b) @ Wu2b + b, transposed out.
// grid = B*64 blocks; block handles 64 rows of the 4096 (v,n) rows for one b.
// ---------------------------------------------------------------------------
__global__ __launch_bounds__(256)
void kB(const float* __restrict__ ws_sigma, const float* __restrict__ Wu2a,
        const float* __restrict__ bu2a, const float* __restrict__ Wu2b,
        const float* __restrict__ bu2b, float* __restrict__ out_U2)
{
  const int blk = blockIdx.x;
  const int b = blk >> 6;
  const int rbase = (blk & 63) * 64;
  const int tid = threadIdx.x;
  const int lane = tid & 31, wid = tid >> 5;
  const int half = lane >> 4, li = lane & 15, k2 = half * 2;

  __shared__ float s_T1[64 * 33];

  const float* S = ws_sigma + (size_t)b * 256 * 4096;  // S[m][row]

  // stage 1: T1 = relu(A(64x256) @ Wu2a(256x32) + b); 4x2 tiles, 1 per wave
  {
    int tm = wid >> 1, tn = wid & 1;
    v8f acc = {};
    for (int kb = 0; kb < 256; kb += 4) {
      v2f a, bb;
      a.x  = S[(size_t)(kb + k2 + 0) * 4096 + rbase + tm * 16 + li];
      a.y  = S[(size_t)(kb + k2 + 1) * 4096 + rbase + tm * 16 + li];
      bb.x = Wu2a[(kb + k2 + 0) * 32 + tn * 16 + li];
      bb.y = Wu2a[(kb + k2 + 1) * 32 + tn * 16 + li];
      acc = wmma4(a, bb, acc);
    }
    float bias = bu2a[tn * 16 + li];
    for (int r = 0; r < 8; ++r)
      s_T1[(tm * 16 + r + 8 * half) * 33 + tn * 16 + li] = frelu(acc[r] + bias);
  }
  __syncthreads();

  // stage 2: T2 = T1(64x32) @ Wu2b(32x60->64) + b; 4x4 tiles, 2 per wave
  for (int t8 = 0; t8 < 2; ++t8) {
    int tile = wid * 2 + t8, tm = tile >> 2, tn = tile & 3;
    int j = tn * 16 + li;
    v8f acc = {};
    for (int kb = 0; kb < 32; kb += 4) {
      v2f a, bb;
      a.x  = s_T1[(tm * 16 + li) * 33 + kb + k2 + 0];
      a.y  = s_T1[(tm * 16 + li) * 33 + kb + k2 + 1];
      bb.x = (j < 60) ? Wu2b[(kb + k2 + 0) * 60 + j] : 0.0f;
      bb.y = (j < 60) ? Wu2b[(kb + k2 + 1) * 60 + j] : 0.0f;
      acc = wmma4(a, bb, acc);
    }
    if (j < 60) {
      float bias = bu2b[j];
      for (int r = 0; r < 8; ++r) {
        int row = rbase + tm * 16 + r + 8 * half;
        out_U2[(size_t)b * 60 * 4096 + (size_t)j * 4096 + row] = acc[r] + bias;
      }
    }
  }
}

// ---------------------------------------------------------------------------
// Kernel C1: mean_all, then relu-hidden layer Hcat = [h1 | h2] into ws_H.
// grid = B*8 blocks; each block does a 64-wide slice of the 512 columns.
// ---------------------------------------------------------------------------
__global__ __launch_bounds__(256)
void kC1(const float* __restrict__ x, const float* __restrict__ ws_mean,
         const float* __restrict__ Wm_in, const float* __restrict__ bm_in,
         const float* __restrict__ Wm_raw, const float* __restrict__ bm_raw,
         float* __restrict__ ws_ma, float* __restrict__ ws_H)
{
  const int blk = blockIdx.x;
  const int b = blk >> 3;
  const int ns = (blk & 7) * 64;
  const int tid = threadIdx.x;
  const int lane = tid & 31, wid = tid >> 5;
  const int half = lane >> 4, li = lane & 15, k2 = half * 2;

  __shared__ float s_ma[64];

  const float* wm = ws_mean + (size_t)b * 256 * 64;
  if (tid < 64) {
    float s = 0.0f;
    for (int mm = 0; mm < 256; ++mm) s += wm[mm * 64 + tid];
    float v = s * (1.0f / 256.0f);
    s_ma[tid] = v;
    ws_ma[b * 64 + tid] = v;   // all 8 slice-blocks write identical values
  }
  __syncthreads();

  float* H = ws_H + (size_t)b * 64 * 1024;

  // h1 slice: (64x256) @ Wm_in(256x512), K=256; 16 tiles, 2 per wave
  for (int t8 = 0; t8 < 2; ++t8) {
    int tile = wid * 2 + t8, tm = tile >> 2, tn = tile & 3;
    int n = ns + tn * 16 + li, d = tm * 16 + li;
    v8f acc = {};
    for (int kb = 0; kb < 256; kb += 4) {
      v2f a, bb;
      a.x  = wm[(kb + k2 + 0) * 64 + d] - s_ma[d];
      a.y  = wm[(kb + k2 + 1) * 64 + d] - s_ma[d];
      bb.x = Wm_in[(kb + k2 + 0) * 512 + n];
      bb.y = Wm_in[(kb + k2 + 1) * 512 + n];
      acc = wmma4(a, bb, acc);
    }
    float bias = bm_in[n];
    for (int r = 0; r < 8; ++r)
      H[(tm * 16 + r + 8 * half) * 1024 + n] = frelu(acc[r] + bias);
  }

  // h2 slice: (64x3072) @ Wm_raw(3072x512), K=3072
  const float* xb = x + (size_t)b * 3072 * 64;
  for (int t8 = 0; t8 < 2; ++t8) {
    int tile = wid * 2 + t8, tm = tile >> 2, tn = tile & 3;
    int n = ns + tn * 16 + li, d = tm * 16 + li;
    v8f acc = {};
    for (int kb = 0; kb < 3072; kb += 4) {
      v2f a, bb;
      a.x  = xb[(size_t)(kb + k2 + 0) * 64 + d] - s_ma[d];
      a.y  = xb[(size_t)(kb + k2 + 1) * 64 + d] - s_ma[d];
      bb.x = Wm_raw[(size_t)(kb + k2 + 0) * 512 + n];
      bb.y = Wm_raw[(size_t)(kb + k2 + 1) * 512 + n];
      acc = wmma4(a, bb, acc);
    }
    float bias = bm_raw[n];
    for (int r = 0; r < 8; ++r)
      H[(tm * 16 + r + 8 * half) * 1024 + 512 + n] = frelu(acc[r] + bias);
  }
}

// ---------------------------------------------------------------------------
// Kernel C2: h = Hcat(64x1024) @ Wm_out(1024x60) + b; preds[b][p][d].
// grid = B blocks.
// ---------------------------------------------------------------------------
__global__ __launch_bounds__(256)
void kC2(const float* __restrict__ ws_H, const float* __restrict__ ws_ma,
         const float* __restrict__ Wm_out, const float* __restrict__ bm_out,
         const float* __restrict__ weight, float* __restrict__ out_preds)
{
  const int b = blockIdx.x;
  const int tid = threadIdx.x;
  const int lane = tid & 31, wid = tid >> 5;
  const int half = lane >> 4, li = lane & 15, k2 = half * 2;

  const float* H = ws_H + (size_t)b * 64 * 1024;

  for (int t8 = 0; t8 < 2; ++t8) {
    int tile = wid * 2 + t8, tm = tile >> 2, tn = tile & 3;
    int p = tn * 16 + li;
    v8f acc = {};
    for (int kb = 0; kb < 1024; kb += 4) {
      v2f a, bb;
      a.x  = H[(tm * 16 + li) * 1024 + kb + k2 + 0];
      a.y  = H[(tm * 16 + li) * 1024 + kb + k2 + 1];
      bb.x = (p < 60) ? Wm_out[(kb + k2 + 0) * 60 + p] : 0.0f;
      bb.y = (p < 60) ? Wm_out[(kb + k2 + 1) * 60 + p] : 0.0f;
      acc = wmma4(a, bb, acc);
    }
    if (p < 60) {
      float bias = bm_out[p];
      for (int r = 0; r < 8; ++r) {
        int d = tm * 16 + r + 8 * half;
        float v = (acc[r] + bias) * weight[d] + ws_ma[b * 64 + d] * weight[64 + d];
        out_preds[(size_t)b * 60 * 64 + p * 64 + d] = v;
      }
    }
  }
}

// ---------------------------------------------------------------------------
extern "C" void kernel_launch(void* const* d_in, const int* in_sizes, int n_in,
                              void* d_out, int out_size, void* d_ws, size_t ws_size,
                              hipStream_t stream)
{
  (void)in_sizes; (void)n_in; (void)out_size; (void)ws_size;
  const float* x      = (const float*)d_in[0];
  const float* WU1    = (const float*)d_in[1];
  const float* bU1    = (const float*)d_in[2];
  const float* WU2    = (const float*)d_in[3];
  const float* bU2    = (const float*)d_in[4];
  const float* Wm_in  = (const float*)d_in[5];
  const float* bm_in  = (const float*)d_in[6];
  const float* Wm_raw = (const float*)d_in[7];
  const float* bm_raw = (const float*)d_in[8];
  const float* Wm_out = (const float*)d_in[9];
  const float* bm_out = (const float*)d_in[10];
  const float* Wu2a   = (const float*)d_in[11];
  const float* bu2a   = (const float*)d_in[12];
  const float* Wu2b   = (const float*)d_in[13];
  const float* bu2b   = (const float*)d_in[14];
  const float* weight = (const float*)d_in[15];

  float* out = (float*)d_out;
  float* out_xnorm = out;                                    // 32*3072*64
  float* out_preds = out + 6291456;                          // 32*60*64
  float* out_U2    = out + 6291456 + 122880;                 // 32*60*4096
  float* out_sign  = out + 6291456 + 122880 + 7864320;       // 32*256*4096

  float* ws = (float*)d_ws;
  float* ws_sigma = ws;                                      // 33,554,432 f
  float* ws_mean  = ws + 33554432;                           // 524,288 f
  float* ws_ma    = ws + 33554432 + 524288;                  // 2,048 f
  float* ws_H     = ws + 33554432 + 524288 + 2048;           // 2,097,152 f

  kA <<<8192, 256, 0, stream>>>(x, WU1, bU1, WU2, bU2,
                                out_xnorm, out_sign, ws_sigma, ws_mean);
  kB <<<2048, 256, 0, stream>>>(ws_sigma, Wu2a, bu2a, Wu2b, bu2b, out_U2);
  kC1<<<256,  256, 0, stream>>>(x, ws_mean, Wm_in, bm_in, Wm_raw, bm_raw,
                                ws_ma, ws_H);
  kC2<<<32,   256, 0, stream>>>(ws_H, ws_ma, Wm_out, bm_out, weight, out_preds);
}